// UserEncoder_89945205113092
// MI455X (gfx1250) — compile-verified
//
#include <hip/hip_runtime.h>
#include <hip/hip_bf16.h>
#include <math.h>

#define E 128
#define L 64
#define NEG_INF_F (-100000000.0f)
#define W0T_STRIDE 130   // padded row stride (floats) to avoid LDS bank conflicts

typedef __attribute__((ext_vector_type(2))) float v2f;
typedef __attribute__((ext_vector_type(8))) float v8f;

// ---------------------------------------------------------------------------
// Phase 1: per (batch, branch) attention-pooled feature.
// block = 128 threads (4 waves), grid = (B, 3)
// ---------------------------------------------------------------------------
__global__ __launch_bounds__(128) void branch_kernel(
    const float* __restrict__ user_embs, const float* __restrict__ item_embs,
    const float* __restrict__ w_uu, const float* __restrict__ w_uiu,
    const float* __restrict__ w_uui,
    const int* __restrict__ uid, const int* __restrict__ nbr,
    const int* __restrict__ uiu, const int* __restrict__ uui,
    float* __restrict__ feat_out, int B)
{
    __shared__ float sself[E];
    __shared__ float w2s[E];
    __shared__ float gbuf[L * E];   // 32 KB: staged gathered (summed) rows
    __shared__ float slog[L];
    __shared__ float aw[L];
    __shared__ float wred[8];
    __shared__ float sC0[1];

    const int b      = blockIdx.x;
    const int branch = blockIdx.y;
    const int tid  = threadIdx.x;
    const int wave = tid >> 5;
    const int lane = tid & 31;

    const float* w = (branch == 0) ? w_uu : ((branch == 1) ? w_uiu : w_uui);
    const float scale = (branch == 0) ? 0.5f : (1.0f / 3.0f);

    // self embedding + second-half weights staged in LDS
    const int u = uid[b];
    const float sv  = user_embs[(size_t)u * E + tid];
    const float w2v = w[E + tid];
    sself[tid] = sv;
    w2s[tid]   = w2v;

    // c0 = self.w[:E] + scale * self.w[E:]
    float p = sv * w[tid];
    float q = sv * w2v;
    #pragma unroll
    for (int off = 16; off; off >>= 1) {
        p += __shfl_xor(p, off, 32);
        q += __shfl_xor(q, off, 32);
    }
    if (lane == 0) { wred[wave] = p; wred[4 + wave] = q; }
    __syncthreads();
    if (tid == 0) {
        float s1 = wred[0] + wred[1] + wred[2] + wred[3];
        float s2 = wred[4] + wred[5] + wred[6] + wred[7];
        sC0[0] = s1 + scale * s2;
    }
    __syncthreads();
    const float c0 = sC0[0];

    // Stage A: gather rows, stage in LDS, compute masked leaky logits
    const float wa = w2s[lane * 4 + 0];
    const float wb = w2s[lane * 4 + 1];
    const float wc = w2s[lane * 4 + 2];
    const float wd = w2s[lane * 4 + 3];
    for (int i = 0; i < 16; ++i) {
        const int l = wave * 16 + i;
        float4 g;
        bool masked;
        if (branch == 0) {
            const int idx = nbr[(size_t)b * L + l];
            masked = (idx == 0);
            g = ((const float4*)(user_embs + (size_t)idx * E))[lane];
        } else if (branch == 1) {
            const int iid = uiu[(size_t)b * 2 * L + 0 * L + l];
            const int uu2 = uiu[(size_t)b * 2 * L + 1 * L + l];
            masked = (iid == 0);
            const float4 gi = ((const float4*)(item_embs + (size_t)iid * E))[lane];
            const float4 gu = ((const float4*)(user_embs + (size_t)uu2 * E))[lane];
            g = make_float4(gi.x + gu.x, gi.y + gu.y, gi.z + gu.z, gi.w + gu.w);
        } else {
            const int uu2 = uui[(size_t)b * 2 * L + 0 * L + l];
            const int iid = uui[(size_t)b * 2 * L + 1 * L + l];
            masked = (uu2 == 0);
            const float4 gu = ((const float4*)(user_embs + (size_t)uu2 * E))[lane];
            const float4 gi = ((const float4*)(item_embs + (size_t)iid * E))[lane];
            g = make_float4(gi.x + gu.x, gi.y + gu.y, gi.z + gu.z, gi.w + gu.w);
        }
        ((float4*)(gbuf + l * E))[lane] = g;
        float d = g.x * wa + g.y * wb + g.z * wc + g.w * wd;
        #pragma unroll
        for (int off = 16; off; off >>= 1) d += __shfl_xor(d, off, 32);
        if (lane == 0) {
            float lg = c0 + scale * d;
            lg = (lg > 0.0f) ? lg : 0.01f * lg;     // leaky_relu, slope 0.01
            if (masked) lg += NEG_INF_F;            // mask AFTER activation
            slog[l] = lg;
        }
    }
    __syncthreads();

    // Stage B: softmax over L logits
    float mx = slog[0];
    for (int l = 1; l < L; ++l) mx = fmaxf(mx, slog[l]);
    float sm = 0.0f;
    for (int l = 0; l < L; ++l) sm += expf(slog[l] - mx);
    if (tid < L) aw[tid] = expf(slog[tid] - mx) / sm;
    __syncthreads();

    // Stage C: weighted sum over staged rows; out = relu(scale*(self + sum a*g))
    float acc = 0.0f;
    for (int l = 0; l < L; ++l) acc += aw[l] * gbuf[l * E + tid];
    float o = scale * (sself[tid] + acc);
    o = fmaxf(o, 0.0f);
    feat_out[((size_t)branch * B + b) * E + tid] = o;
}

// ---------------------------------------------------------------------------
// Zero the three score accumulators
// ---------------------------------------------------------------------------
__global__ void zero_scores(float* __restrict__ scores)
{
    if (threadIdx.x < 4) scores[threadIdx.x] = 0.0f;
}

// ---------------------------------------------------------------------------
// Phase 2: scores[branch] = sum_rows tanh(feat @ W0 + b0) . w1  via f32 WMMA.
// W0 staged TRANSPOSED (+padded stride) in LDS so each B fragment is one
// contiguous 8-byte ds_load_b64 with no register repacking and no bank
// conflicts: bank = (n*130 + k) % 64 = (2n + k) % 64, distinct per lane.
// block = 256 threads (8 waves, 16 rows each => 128 rows/block);
// grid = 3 * (B/128) blocks.
// ---------------------------------------------------------------------------
__global__ __launch_bounds__(256) void inter_kernel(
    const float* __restrict__ feat,   // [3*B, E]
    const float* __restrict__ W0,     // [E, 64] row-major
    const float* __restrict__ b0,     // [64]
    const float* __restrict__ w1,     // [64]
    float* __restrict__ scores, int B)
{
    __shared__ float W0t[64 * W0T_STRIDE];   // transposed: W0t[n*STRIDE + k] = W0[k][n]
    __shared__ float b0s[64];
    __shared__ float w1s[64];
    __shared__ float wsum[8];

    const int tid = threadIdx.x;
    const int blocksPerBranch = B / 128;
    const int branch = blockIdx.x / blocksPerBranch;
    const int blkIn  = blockIdx.x % blocksPerBranch;

    // stage W0 transposed; 8192 elements / 256 threads = 32 each
    for (int i = tid; i < E * 64; i += 256) {
        const int k = i >> 6;          // 0..127
        const int n = i & 63;          // 0..63
        W0t[n * W0T_STRIDE + k] = W0[i];
    }
    if (tid < 64) { b0s[tid] = b0[tid]; w1s[tid] = w1[tid]; }
    __syncthreads();

    const int wave = tid >> 5;
    const int lane = tid & 31;
    const int m  = lane & 15;        // row-in-tile (A) / col-in-tile (B,C)
    const int hi = lane >> 4;        // lane half selects K pair

    const size_t row = (size_t)branch * B + (size_t)blkIn * 128 + wave * 16 + m;
    const float* Xrow = feat + row * E;

    // per-lane base pointers into the 4 transposed B columns (n = nt*16 + m)
    const float* Bp0 = W0t + (0 * 16 + m) * W0T_STRIDE;
    const float* Bp1 = W0t + (1 * 16 + m) * W0T_STRIDE;
    const float* Bp2 = W0t + (2 * 16 + m) * W0T_STRIDE;
    const float* Bp3 = W0t + (3 * 16 + m) * W0T_STRIDE;

    v8f acc0 = {}, acc1 = {}, acc2 = {}, acc3 = {};
    for (int kk = 0; kk < E; kk += 4) {
        const int ka = kk + 2 * hi;
        // A fragment: lane holds A[m][ka], A[m][ka+1] (shared by all 4 N-tiles)
        v2f a = *(const v2f*)(Xrow + ka);
        // B fragments: contiguous float2 loads from transposed LDS
        v2f bf0 = *(const v2f*)(Bp0 + ka);
        v2f bf1 = *(const v2f*)(Bp1 + ka);
        v2f bf2 = *(const v2f*)(Bp2 + ka);
        v2f bf3 = *(const v2f*)(Bp3 + ka);
        acc0 = __builtin_amdgcn_wmma_f32_16x16x4_f32(false, a, false, bf0, (short)0, acc0, false, false);
        acc1 = __builtin_amdgcn_wmma_f32_16x16x4_f32(false, a, false, bf1, (short)0, acc1, false, false);
        acc2 = __builtin_amdgcn_wmma_f32_16x16x4_f32(false, a, false, bf2, (short)0, acc2, false, false);
        acc3 = __builtin_amdgcn_wmma_f32_16x16x4_f32(false, a, false, bf3, (short)0, acc3, false, false);
    }

    // Epilogue: sum over all cells of tanh(acc + b0[n]) * w1[n].
    // Only the N index matters per lane (N = nt*16 + m), robust to M half-swap.
    float psum = 0.0f;
    {
        v8f accs[4] = { acc0, acc1, acc2, acc3 };
        #pragma unroll
        for (int nt = 0; nt < 4; ++nt) {
            const int n = nt * 16 + m;
            const float bb = b0s[n];
            const float ww = w1s[n];
            #pragma unroll
            for (int r = 0; r < 8; ++r) {
                psum += tanhf(accs[nt][r] + bb) * ww;
            }
        }
    }
    #pragma unroll
    for (int off = 16; off; off >>= 1) psum += __shfl_xor(psum, off, 32);
    if (lane == 0) wsum[wave] = psum;
    __syncthreads();
    if (tid == 0) {
        float s = 0.0f;
        #pragma unroll
        for (int i = 0; i < 8; ++i) s += wsum[i];
        atomicAdd(&scores[branch], s);
    }
}

// ---------------------------------------------------------------------------
// Phase 3: softmax over the 3 (mean) scores, relu-combine the branch features.
// ---------------------------------------------------------------------------
__global__ __launch_bounds__(256) void combine_kernel(
    const float* __restrict__ feat, const float* __restrict__ scores,
    float* __restrict__ out, int B)
{
    const size_t i = (size_t)blockIdx.x * 256 + threadIdx.x;
    const size_t N = (size_t)B * E;
    if (i >= N) return;
    const float inv = 1.0f / (float)B;
    const float s0 = scores[0] * inv;
    const float s1 = scores[1] * inv;
    const float s2 = scores[2] * inv;
    const float mx = fmaxf(s0, fmaxf(s1, s2));
    const float e0 = expf(s0 - mx);
    const float e1 = expf(s1 - mx);
    const float e2 = expf(s2 - mx);
    const float den = e0 + e1 + e2;
    const float v = (e0 * feat[i] + e1 * feat[N + i] + e2 * feat[2 * N + i]) / den;
    out[i] = fmaxf(v, 0.0f);
}

// ---------------------------------------------------------------------------
extern "C" void kernel_launch(void* const* d_in, const int* in_sizes, int n_in,
                              void* d_out, int out_size, void* d_ws, size_t ws_size,
                              hipStream_t stream) {
    const float* user_embs = (const float*)d_in[0];
    const float* item_embs = (const float*)d_in[1];
    const float* w_uu      = (const float*)d_in[2];
    const float* w_uiu     = (const float*)d_in[3];
    const float* w_uui     = (const float*)d_in[4];
    const float* W0        = (const float*)d_in[5];
    const float* b0        = (const float*)d_in[6];
    const float* w1        = (const float*)d_in[7];
    const int*   uid       = (const int*)d_in[8];
    const int*   nbr       = (const int*)d_in[9];
    const int*   uiu       = (const int*)d_in[10];
    const int*   uui       = (const int*)d_in[11];

    const int B = in_sizes[8];               // 4096

    float* ws_feat   = (float*)d_ws;                          // 3*B*E floats
    float* ws_scores = ws_feat + (size_t)3 * B * E;           // 3 floats (+pad)

    zero_scores<<<1, 32, 0, stream>>>(ws_scores);

    dim3 g1(B, 3);
    branch_kernel<<<g1, 128, 0, stream>>>(user_embs, item_embs,
                                          w_uu, w_uiu, w_uui,
                                          uid, nbr, uiu, uui,
                                          ws_feat, B);

    inter_kernel<<<3 * (B / 128), 256, 0, stream>>>(ws_feat, W0, b0, w1,
                                                    ws_scores, B);

    const int total = B * E;
    combine_kernel<<<(total + 255) / 256, 256, 0, stream>>>(ws_feat, ws_scores,
                                                            (float*)d_out, B);
}